// VectorQuantizer_86775519248937
// MI455X (gfx1250) — compile-verified
//
#include <hip/hip_runtime.h>

// Problem constants (from reference)
#define NUM_EMB   8192
#define EMB_DIM   64
#define N_ROWS    16384                 // 16 * 1024 flattened z vectors
#define ZQ_ELEMS  (N_ROWS * EMB_DIM)    // 1048576
#define IDX_OFF   ZQ_ELEMS              // indices start here in d_out (as float)
#define LOSS_OFF  (ZQ_ELEMS + N_ROWS)   // scalar vq_loss slot
#define NTILES    (NUM_EMB / 16)        // 512
#define LDS_STRIDE 68                   // padded row stride (floats) -> bank-conflict-free

typedef __attribute__((ext_vector_type(2))) float v2f;
typedef __attribute__((ext_vector_type(8))) float v8f;

// ---------------------------------------------------------------------------
// Kernel 1: codebook squared norms  ||e_k||^2  (8192 rows of 64 f32)
// ---------------------------------------------------------------------------
__global__ void vq_norms(const float* __restrict__ cb, float* __restrict__ cbnorm) {
    int row = blockIdx.x * blockDim.x + threadIdx.x;   // 0..8191
    const float4* p = (const float4*)(cb + row * EMB_DIM);
    float s = 0.f;
#pragma unroll
    for (int i = 0; i < EMB_DIM / 4; ++i) {
        float4 v = p[i];
        s += v.x * v.x + v.y * v.y + v.z * v.z + v.w * v.w;
    }
    cbnorm[row] = s;
}

// ---------------------------------------------------------------------------
// Kernel 2: fused distance GEMM (f32 WMMA) + argmin + gather + loss partials
//   128 blocks x 128 threads (4 waves). Wave w owns TWO 16-row M-tiles
//   (32 z rows) -> two independent WMMA accumulator chains per B-tile.
//   The block streams the codebook through a double-buffered LDS tile filled
//   with CDNA5 async global->LDS copies (ASYNCcnt), shared by all 4 waves.
// ---------------------------------------------------------------------------
__global__ __launch_bounds__(128) void vq_main(
        const float* __restrict__ z, const float* __restrict__ cb,
        const float* __restrict__ cbnorm, float* __restrict__ out,
        float* __restrict__ partials) {
    __shared__ float smem[2][16 * LDS_STRIDE + 8];

    const int tid  = threadIdx.x;
    const int lane = tid & 31;
    const int wave = tid >> 5;
    const int m    = lane & 15;   // A: row-in-tile, B: codebook entry in tile
    const int hi   = lane >> 4;   // K-half bit of WMMA f32 16x16x4 layout
    const int rowBase0 = blockIdx.x * 128 + wave * 32;      // M-tile 0
    const int rowBase1 = rowBase0 + 16;                     // M-tile 1

    // Staging geometry: tile = 16 entries x 64 f32 = 4KB = 256 x b128 chunks.
    // Thread t moves chunks t and t+128 (two async b128 instrs per wave/tile).
    const int e0 = tid >> 4;               // entry for chunk `tid`       (0..7)
    const int c0 = (tid & 15) * 4;         // float col for chunk
    const int e1 = (tid + 128) >> 4;       // entry for chunk `tid+128`   (8..15)
    const unsigned ldsOff0[2] = {
        (unsigned)(size_t)&smem[0][e0 * LDS_STRIDE + c0],
        (unsigned)(size_t)&smem[1][e0 * LDS_STRIDE + c0] };
    const unsigned ldsOff1[2] = {
        (unsigned)(size_t)&smem[0][e1 * LDS_STRIDE + c0],
        (unsigned)(size_t)&smem[1][e1 * LDS_STRIDE + c0] };

    auto issue_tile = [&](int tile, int buf) {
        int n0 = (tile < NTILES) ? tile * 16 : 0;   // clamp: tail dummies re-read tile 0
        const float* g0 = cb + (n0 + e0) * EMB_DIM + c0;
        const float* g1 = cb + (n0 + e1) * EMB_DIM + c0;
        asm volatile("global_load_async_to_lds_b128 %0, %1, off"
                     :: "v"(ldsOff0[buf]), "v"(g0) : "memory");
        asm volatile("global_load_async_to_lds_b128 %0, %1, off"
                     :: "v"(ldsOff1[buf]), "v"(g1) : "memory");
    };

    // ---- Preload A fragments for all 16 K-steps (K = 4s + 2*hi + {0,1}) ----
    v2f a0[16], a1[16];
    {
        const float* zr0 = z + (rowBase0 + m) * EMB_DIM + 2 * hi;
        const float* zr1 = z + (rowBase1 + m) * EMB_DIM + 2 * hi;
#pragma unroll
        for (int s = 0; s < 16; ++s) {
            a0[s] = *(const v2f*)(zr0 + 4 * s);
            a1[s] = *(const v2f*)(zr1 + 4 * s);
        }
    }

    float bv0[8], bv1[8];
    int   bi0[8], bi1[8];
#pragma unroll
    for (int r = 0; r < 8; ++r) {
        bv0[r] = 3.402823466e38f; bi0[r] = 0;
        bv1[r] = 3.402823466e38f; bi1[r] = 0;
    }

    // ---- Prologue: fill both buffers -------------------------------------
    issue_tile(0, 0);
    issue_tile(1, 1);

    // ---- Main loop over 512 codebook tiles -------------------------------
    for (int t = 0; t < NTILES; ++t) {
        // Outstanding: tile t (<=2/wave) + tile t+1 (2/wave); in-order retire
        // means asynccnt<=2 guarantees tile t landed in LDS.
        asm volatile("s_wait_asynccnt 0x2" ::: "memory");
        __syncthreads();                    // all waves' portions visible

        const float* brow = &smem[t & 1][0] + m * LDS_STRIDE + 2 * hi;

        v8f acc0 = {}, acc1 = {};
#pragma unroll
        for (int s = 0; s < 16; ++s) {
            v2f b = *(const v2f*)(brow + 4 * s);          // ds_load_b64
            acc0 = __builtin_amdgcn_wmma_f32_16x16x4_f32(
                false, a0[s], false, b, (short)0, acc0, false, false);
            acc1 = __builtin_amdgcn_wmma_f32_16x16x4_f32(
                false, a1[s], false, b, (short)0, acc1, false, false);
        }

        const int   nIdx = t * 16 + m;
        const float cbn  = cbnorm[nIdx];
#pragma unroll
        for (int r = 0; r < 8; ++r) {
            float d0 = __builtin_fmaf(-2.f, acc0[r], cbn);  // ||e||^2 - 2 z.e
            float d1 = __builtin_fmaf(-2.f, acc1[r], cbn);
            if (d0 < bv0[r]) { bv0[r] = d0; bi0[r] = nIdx; } // strict <: first idx wins
            if (d1 < bv1[r]) { bv1[r] = d1; bi1[r] = nIdx; }
        }

        __syncthreads();                    // everyone done reading buf[t&1]
        issue_tile(t + 2, t & 1);           // safe to overwrite it now
    }

    // ---- Cross-lane argmin within each 16-lane half (rows r / r+8) -------
#pragma unroll
    for (int r = 0; r < 8; ++r) {
#pragma unroll
        for (int off = 8; off >= 1; off >>= 1) {
            float ov0 = __shfl_xor(bv0[r], off, 32);
            int   oi0 = __shfl_xor(bi0[r], off, 32);
            if (ov0 < bv0[r] || (ov0 == bv0[r] && oi0 < bi0[r])) { bv0[r] = ov0; bi0[r] = oi0; }
            float ov1 = __shfl_xor(bv1[r], off, 32);
            int   oi1 = __shfl_xor(bi1[r], off, 32);
            if (ov1 < bv1[r] || (ov1 == bv1[r] && oi1 < bi1[r])) { bv1[r] = ov1; bi1[r] = oi1; }
        }
    }

    // ---- Broadcast: lane L needs winner of row m = L%16 in each group ----
    int myidx0 = 0, myidx1 = 0;
#pragma unroll
    for (int r = 0; r < 8; ++r) {
        int v00 = __shfl(bi0[r], 0, 32);   // rows 0..7  live in lanes 0..15
        int v01 = __shfl(bi0[r], 16, 32);  // rows 8..15 live in lanes 16..31
        int v10 = __shfl(bi1[r], 0, 32);
        int v11 = __shfl(bi1[r], 16, 32);
        if (m == r)     { myidx0 = v00; myidx1 = v10; }
        if (m == r + 8) { myidx0 = v01; myidx1 = v11; }
    }

    const int grow0 = rowBase0 + m;
    const int grow1 = rowBase1 + m;
    if (hi == 0) {
        out[IDX_OFF + grow0] = (float)myidx0;  // indices as float values
        out[IDX_OFF + grow1] = (float)myidx1;
    }

    // ---- Gather z_q = codebook[idx] and fused (z_q - z)^2 partial --------
    float sq = 0.f;
#pragma unroll
    for (int g = 0; g < 2; ++g) {
        const int grow  = g ? grow1 : grow0;
        const int myidx = g ? myidx1 : myidx0;
        const float4* src = (const float4*)(cb + myidx * EMB_DIM + hi * 32);
        const float4* zr4 = (const float4*)(z + grow * EMB_DIM + hi * 32);
        float4*       dst = (float4*)(out + grow * EMB_DIM + hi * 32);
#pragma unroll
        for (int i = 0; i < 8; ++i) {
            float4 c = src[i], zz = zr4[i];
            dst[i] = c;  // z_q_ste == z_q numerically
            float dx = c.x - zz.x, dy = c.y - zz.y, dz = c.z - zz.z, dw = c.w - zz.w;
            sq += dx * dx + dy * dy + dz * dz + dw * dw;
        }
    }
#pragma unroll
    for (int off = 16; off >= 1; off >>= 1) sq += __shfl_xor(sq, off, 32);
    if (lane == 0) partials[blockIdx.x * 4 + wave] = sq;   // 512 partials total
}

// ---------------------------------------------------------------------------
// Kernel 3: deterministic fixed-tree reduction of 512 partials -> vq_loss
//   vq_loss = (1 + 0.25) * mean((z_q - z)^2)
// ---------------------------------------------------------------------------
__global__ void vq_loss_reduce(const float* __restrict__ partials,
                               float* __restrict__ loss_out) {
    __shared__ float sm[256];
    int t = threadIdx.x;
    sm[t] = partials[t] + partials[t + 256];
    __syncthreads();
#pragma unroll
    for (int off = 128; off >= 1; off >>= 1) {
        if (t < off) sm[t] += sm[t + off];
        __syncthreads();
    }
    if (t == 0) *loss_out = 1.25f * sm[0] / (float)ZQ_ELEMS;
}

// ---------------------------------------------------------------------------
extern "C" void kernel_launch(void* const* d_in, const int* in_sizes, int n_in,
                              void* d_out, int out_size, void* d_ws, size_t ws_size,
                              hipStream_t stream) {
    const float* z  = (const float*)d_in[0];   // (16,1024,64) f32
    const float* cb = (const float*)d_in[1];   // (8192,64)    f32
    float* out = (float*)d_out;
    float* wsf = (float*)d_ws;
    float* cbnorm   = wsf;              // 8192 floats
    float* partials = wsf + NUM_EMB;    // 512 floats

    vq_norms<<<NUM_EMB / 128, 128, 0, stream>>>(cb, cbnorm);
    vq_main<<<N_ROWS / 128, 128, 0, stream>>>(z, cb, cbnorm, out, partials);
    vq_loss_reduce<<<1, 256, 0, stream>>>(partials, out + LOSS_OFF);
}